// ResidueReadout_91190745629085
// MI455X (gfx1250) — compile-verified
//
#include <hip/hip_runtime.h>
#include <hip/hip_bf16.h>

// ---------------------------------------------------------------------------
// ResidueReadout: segmented mean over atoms->residues, then scatter residues
// into a zero-padded [P, max_len, 128] tensor.
//
// Memory-bound problem (~486 MB read, 52 MB write => ~23 us at 23.3 TB/s).
// The segment-mean is expressed as D = A x B + C with V_WMMA_F32_16X16X4_F32:
//   A (16x4 f32)  = per-tile segment-indicator weights (1/size or 0)
//   B (4x16 f32)  = streamed node-feature fragments
//   C/D (16x16)   = 16 residues x 16 features accumulator (8 tiles for D=128)
// ---------------------------------------------------------------------------

typedef __attribute__((ext_vector_type(2))) float v2f;
typedef __attribute__((ext_vector_type(8))) float v8f;

#define DFEAT 128

// ---------------- exclusive prefix scan (single workgroup, chunked) --------
__global__ void __launch_bounds__(1024)
scan_sizes_kernel(const int* __restrict__ sizes, int n, int* __restrict__ starts) {
    __shared__ int buf[1024];
    __shared__ int carry_s;
    const int tid = threadIdx.x;
    if (tid == 0) { carry_s = 0; starts[0] = 0; }
    __syncthreads();
    for (int base = 0; base < n; base += 1024) {
        const int i = base + tid;
        int v = (i < n) ? sizes[i] : 0;
        buf[tid] = v;
        __syncthreads();
        // Hillis-Steele inclusive scan over 1024 elements
        #pragma unroll
        for (int off = 1; off < 1024; off <<= 1) {
            int t = (tid >= off) ? buf[tid - off] : 0;
            __syncthreads();
            buf[tid] += t;
            __syncthreads();
        }
        const int incl = buf[tid];
        const int carry = carry_s;
        if (i < n) starts[i + 1] = carry + incl;
        __syncthreads();
        if (tid == 1023) carry_s = carry + incl;
        __syncthreads();
    }
}

// ---------------- residue -> flat output row map ---------------------------
__global__ void __launch_bounds__(256)
map_rows_kernel(const int* __restrict__ pep_start, int P, int max_len,
                int* __restrict__ res_to_out) {
    const int p = blockIdx.x * blockDim.x + threadIdx.x;
    if (p >= P) return;
    const int s = pep_start[p];
    const int e = pep_start[p + 1];
    const int rowbase = p * max_len;
    for (int r = s; r < e; ++r) res_to_out[r] = rowbase + (r - s);
}

// ---------------- zero-fill padded output (float4 stores) ------------------
__global__ void __launch_bounds__(256)
zero_out_kernel(float4* __restrict__ out, long n4) {
    long i = (long)blockIdx.x * blockDim.x + threadIdx.x;
    const long stride = (long)gridDim.x * blockDim.x;
    const float4 z = make_float4(0.f, 0.f, 0.f, 0.f);
    for (; i < n4; i += stride) out[i] = z;
}

// ---------------- main WMMA segmented-mean kernel --------------------------
// One wave32 per tile of 16 consecutive residues. Streams the tile's atom
// rows once; 8 accumulators of v_wmma_f32_16x16x4_f32 cover D=128 features.
__global__ void __launch_bounds__(256)
residue_mean_wmma_kernel(const float* __restrict__ X,
                         const int* __restrict__ res_start,
                         const int* __restrict__ res_to_out,
                         float* __restrict__ out,
                         int R, int N) {
    const int lane      = threadIdx.x & 31;
    const int wave      = threadIdx.x >> 5;
    const int tile      = blockIdx.x * (blockDim.x >> 5) + wave;
    const int r0        = tile * 16;
    if (r0 >= R) return;                       // wave-uniform exit
    const int nres      = min(16, R - r0);

    const int m = lane & 15;                   // residue row within tile
    int s0 = 0, s1 = 0;                        // atom range of residue m
    if (m < nres) { s0 = res_start[r0 + m]; s1 = res_start[r0 + m + 1]; }
    const float inv = (s1 > s0) ? (1.0f / (float)(s1 - s0)) : 0.0f;

    const int ab = res_start[r0];
    const int ae = res_start[r0 + nres];
    const int khalf = (lane >> 4) << 1;        // lanes 0-15 -> K{0,1}; 16-31 -> K{2,3}
    const int n = m;                           // feature column within 16-wide tile

    v8f acc[8] = {};

    for (int k0 = ab; k0 < ae; k0 += 4) {
        const int g0 = k0 + khalf;
        const int g1 = g0 + 1;
        // A fragment: indicator * (1/size), zero outside this residue's range
        v2f A;
        A.x = (g0 >= s0 && g0 < s1) ? inv : 0.0f;
        A.y = (g1 >= s0 && g1 < s1) ? inv : 0.0f;

        const float* p0 = X + (size_t)min(g0, N - 1) * DFEAT + n;
        const float* p1 = X + (size_t)min(g1, N - 1) * DFEAT + n;

        // Prefetch one K-block (4 rows) ahead -> global_prefetch_b8
        __builtin_prefetch(X + (size_t)min(g0 + 4, N - 1) * DFEAT + n, 0, 1);

        #pragma unroll
        for (int ft = 0; ft < 8; ++ft) {
            v2f B;
            B.x = p0[ft * 16];
            B.y = p1[ft * 16];
            // D = A(16x4) * B(4x16) + C ; args: (neg_a, A, neg_b, B, c_mod, C, reuse_a, reuse_b)
            acc[ft] = __builtin_amdgcn_wmma_f32_16x16x4_f32(
                false, A, false, B, (short)0, acc[ft], false, false);
        }
    }

    // Scatter: VGPR j of C/D holds M=j (lanes 0-15) and M=j+8 (lanes 16-31)
    const int mbase = (lane >> 4) << 3;        // 0 or 8
    #pragma unroll
    for (int j = 0; j < 8; ++j) {
        const int mm = mbase + j;
        if (mm < nres) {
            const int orow = res_to_out[r0 + mm];
            float* op = out + (size_t)orow * DFEAT + n;
            #pragma unroll
            for (int ft = 0; ft < 8; ++ft) op[ft * 16] = acc[ft][j];
        }
    }
}

// ---------------------------------------------------------------------------
extern "C" void kernel_launch(void* const* d_in, const int* in_sizes, int n_in,
                              void* d_out, int out_size, void* d_ws, size_t ws_size,
                              hipStream_t stream) {
    const float* X        = (const float*)d_in[0];
    const int*   res_size = (const int*)d_in[1];
    const int*   pep_size = (const int*)d_in[2];
    float*       out      = (float*)d_out;

    const int N = in_sizes[0] / DFEAT;   // atoms
    const int R = in_sizes[1];           // residues
    const int P = in_sizes[2];           // peptides
    const int max_len = out_size / (P * DFEAT);

    int* ws         = (int*)d_ws;
    int* res_start  = ws;                        // R+1 ints
    int* pep_start  = res_start + (R + 1);       // P+1 ints
    int* res_to_out = pep_start + (P + 1);       // R ints

    // 1) prefix scans (tiny; single-WGP chunked scans)
    scan_sizes_kernel<<<1, 1024, 0, stream>>>(res_size, R, res_start);
    scan_sizes_kernel<<<1, 1024, 0, stream>>>(pep_size, P, pep_start);

    // 2) residue -> padded-output row map
    map_rows_kernel<<<(P + 255) / 256, 256, 0, stream>>>(pep_start, P, max_len,
                                                         res_to_out);

    // 3) zero-fill padded output (covers padding rows)
    const long n4 = (long)out_size / 4;
    int zblocks = (int)((n4 + 255) / 256);
    if (zblocks > 16384) zblocks = 16384;
    zero_out_kernel<<<zblocks, 256, 0, stream>>>((float4*)d_out, n4);

    // 4) main WMMA segmented mean: one wave per 16-residue tile, 8 waves/block
    const int tiles  = (R + 15) / 16;
    const int blocks = (tiles + 7) / 8;
    residue_mean_wmma_kernel<<<blocks, 256, 0, stream>>>(X, res_start, res_to_out,
                                                         out, R, N);
}